// CrissCrossAttention3D_69063074120028
// MI455X (gfx1250) — compile-verified
//
#include <hip/hip_runtime.h>

// ---------------------------------------------------------------------------
// CrissCross 3D attention for MI455X (gfx1250, wave32, WMMA f32 16x16x4).
// Pipeline: qk-proj GEMM -> per-line logits GEMM (3 dirs) -> softmax(144)
//           -> per-line aggregation GEMM in input-channel space (3 dirs)
//           -> final Wv GEMM + bias + gamma + residual.
// v is never materialized: out = Wv * (sum_i a_i * x[:,n_i]) + bv  (sum a = 1).
// ---------------------------------------------------------------------------

#define NB   2
#define NC   256
#define NE   32
#define NL   48            // H == W == D == 48
#define NSP  110592        // 48*48*48
#define NWD  2304          // 48*48
#define NATT 144

#ifndef USE_ASYNC_COPY
#define USE_ASYNC_COPY 1   // global_load_async_to_lds_b128 path (ASYNCcnt)
#endif

typedef float v2f __attribute__((ext_vector_type(2)));
typedef float v8f __attribute__((ext_vector_type(8)));

// D = A(16x4, f32) x B(4x16, f32) + C(16x16, f32)
__device__ __forceinline__ v8f wmma4(v2f a, v2f b, v8f c) {
  return __builtin_amdgcn_wmma_f32_16x16x4_f32(
      /*neg_a=*/false, a, /*neg_b=*/false, b,
      /*c_mod=*/(short)0, c, /*reuse_a=*/false, /*reuse_b=*/false);
}

// ---------------------------------------------------------------------------
// Kernel 1: q = Wq*x + bq ; k = Wk*x + bk   (stacked 64x256 GEMM, N-tile 64)
// ---------------------------------------------------------------------------
__global__ __launch_bounds__(256) void kproj_qk(
    const float* __restrict__ x,
    const float* __restrict__ Wq, const float* __restrict__ bq,
    const float* __restrict__ Wk, const float* __restrict__ bk,
    float* __restrict__ qo, float* __restrict__ ko)
{
  __shared__ float Wl[64][260];   // pad 260: stride%64==4 banks, 16B aligned
  __shared__ float Xl[256][64];
  __shared__ float bl[64];

  const int b  = blockIdx.y;
  const int n0 = blockIdx.x * 64;
  const int tid = threadIdx.x;

  // weights: 64 rows x 256 cols = 4096 float4
  for (int i = tid; i < 64 * 64; i += 256) {
    int row = i >> 6, col = (i & 63) << 2;
    const float* src = (row < 32) ? (Wq + row * NC + col)
                                  : (Wk + (row - 32) * NC + col);
    *(float4*)&Wl[row][col] = *(const float4*)src;
  }
  if (tid < 64) bl[tid] = (tid < 32) ? bq[tid] : bk[tid - 32];

  // X tile: 256 rows x 64 cols = 4096 float4
  for (int i = tid; i < 256 * 16; i += 256) {
    int row = i >> 4, col = (i & 15) << 2;
    const float* g = x + ((size_t)b * NC + row) * NSP + n0 + col;
#if USE_ASYNC_COPY
    unsigned lds = (unsigned)(unsigned long long)&Xl[row][col];
    asm volatile("global_load_async_to_lds_b128 %0, %1, off"
                 :: "v"(lds), "v"(g) : "memory");
#else
    *(float4*)&Xl[row][col] = *(const float4*)g;
#endif
  }
#if USE_ASYNC_COPY
  asm volatile("s_wait_asynccnt 0x0" ::: "memory");
#endif
  __syncthreads();

  const int wave = tid >> 5, lane = tid & 31, half = lane >> 4, l16 = lane & 15;
  for (int t = 0; t < 2; ++t) {
    const int tile = wave * 2 + t, tm = tile >> 2, tn = tile & 3;
    v8f c; for (int r = 0; r < 8; ++r) c[r] = 0.f;
    for (int kb = 0; kb < 256; kb += 4) {
      const int ka = kb + 2 * half;
      v2f a, bb;
      a.x = Wl[tm * 16 + l16][ka];     a.y = Wl[tm * 16 + l16][ka + 1];
      bb.x = Xl[ka][tn * 16 + l16];    bb.y = Xl[ka + 1][tn * 16 + l16];
      c = wmma4(a, bb, c);
    }
    for (int r = 0; r < 8; ++r) {
      const int m = tm * 16 + r + 8 * half;
      const int col = n0 + tn * 16 + l16;
      const float val = c[r] + bl[m];
      if (m < 32) qo[((size_t)b * NE + m) * NSP + col] = val;
      else        ko[((size_t)b * NE + (m - 32)) * NSP + col] = val;
    }
  }
}

// ---------------------------------------------------------------------------
// Kernel 2: per-line logits  E = Q^T K  (48x48, K=32), dir in {H,W,D}
// att layout: [b][n][144], slots [0,48)=H [48,96)=W [96,144)=D
// ---------------------------------------------------------------------------
__global__ __launch_bounds__(96) void klogits(
    const float* __restrict__ q, const float* __restrict__ k,
    float* __restrict__ att)
{
  __shared__ float Ql[32][48];
  __shared__ float Kl[32][48];

  const int b = blockIdx.y, dir = blockIdx.z, li = blockIdx.x;
  const int c0 = li / NL, c1 = li % NL;
  int n0, stride;
  if      (dir == 0) { n0 = c0 * NL + c1;        stride = NWD; } // H-line @(w,d)
  else if (dir == 1) { n0 = c0 * NWD + c1;       stride = NL;  } // W-line @(h,d)
  else               { n0 = c0 * NWD + c1 * NL;  stride = 1;   } // D-line @(h,w)

  const int tid = threadIdx.x;
  for (int i = tid; i < NE * NL; i += 96) {
    const int e = i / NL, xx = i % NL;
    const size_t o = ((size_t)b * NE + e) * NSP + n0 + (size_t)xx * stride;
    Ql[e][xx] = q[o];
    Kl[e][xx] = k[o];
  }
  __syncthreads();

  const int wave = tid >> 5, lane = tid & 31, half = lane >> 4, l16 = lane & 15;
  const int tm = wave;                  // 3 waves cover 3 row-tiles
  for (int tn = 0; tn < 3; ++tn) {
    v8f c; for (int r = 0; r < 8; ++r) c[r] = 0.f;
    for (int kb = 0; kb < NE; kb += 4) {
      const int ka = kb + 2 * half;
      v2f a, bb;
      a.x = Ql[ka][tm * 16 + l16];   a.y = Ql[ka + 1][tm * 16 + l16];
      bb.x = Kl[ka][tn * 16 + l16];  bb.y = Kl[ka + 1][tn * 16 + l16];
      c = wmma4(a, bb, c);
    }
    for (int r = 0; r < 8; ++r) {
      const int xx = tm * 16 + r + 8 * half;   // output position along line
      const int yy = tn * 16 + l16;            // neighbor along line
      const float val = (dir == 0 && xx == yy) ? -__builtin_inff() : c[r];
      att[(((size_t)b * NSP) + n0 + (size_t)xx * stride) * NATT
          + dir * NL + yy] = val;
    }
  }
}

// ---------------------------------------------------------------------------
// Kernel 3: softmax over 144 per position, one wave32 per position, in place
// ---------------------------------------------------------------------------
__global__ __launch_bounds__(256) void ksoftmax(float* __restrict__ att, int npos)
{
  const int wave = threadIdx.x >> 5, lane = threadIdx.x & 31;
  const int p = blockIdx.x * 8 + wave;
  if (p >= npos) return;
  float* a = att + (size_t)p * NATT;

  float v[5];
  float m = -3.4e38f;
  for (int i = 0; i < 5; ++i) {
    const int idx = lane + 32 * i;
    v[i] = (idx < NATT) ? a[idx] : -__builtin_inff();
    m = fmaxf(m, v[i]);
  }
  for (int off = 16; off; off >>= 1) m = fmaxf(m, __shfl_xor(m, off, 32));
  float sum = 0.f;
  for (int i = 0; i < 5; ++i) { v[i] = __expf(v[i] - m); sum += v[i]; }
  for (int off = 16; off; off >>= 1) sum += __shfl_xor(sum, off, 32);
  const float inv = 1.f / sum;
  for (int i = 0; i < 5; ++i) {
    const int idx = lane + 32 * i;
    if (idx < NATT) a[idx] = v[i] * inv;
  }
}

// ---------------------------------------------------------------------------
// Kernel 4: per-line aggregation  S(c,x) (+)= sum_y X(c,y) * a(x,y)
// launched once per direction (sequential -> race-free RMW on s)
// ---------------------------------------------------------------------------
__global__ __launch_bounds__(256) void kaggr(
    const float* __restrict__ x, const float* __restrict__ att,
    float* __restrict__ s, int dir, int accumulate)
{
  __shared__ float Xl[256][49];   // pad 49: bank-conflict-free column reads
  __shared__ float Al[48][49];

  const int b = blockIdx.y, li = blockIdx.x;
  const int c0 = li / NL, c1 = li % NL;
  int n0, stride;
  if      (dir == 0) { n0 = c0 * NL + c1;       stride = NWD; }
  else if (dir == 1) { n0 = c0 * NWD + c1;      stride = NL;  }
  else               { n0 = c0 * NWD + c1 * NL; stride = 1;   }

  const int tid = threadIdx.x;
  for (int i = tid; i < NC * NL; i += 256) {
    const int row = i / NL, col = i % NL;
    const float* g = x + ((size_t)b * NC + row) * NSP + n0 + (size_t)col * stride;
    __builtin_prefetch(g + (size_t)stride * 8, 0, 0);
    Xl[row][col] = *g;
  }
  for (int i = tid; i < NL * NL; i += 256) {
    const int xx = i / NL, yy = i % NL;
    Al[xx][yy] = att[((size_t)b * NSP + n0 + (size_t)xx * stride) * NATT
                     + dir * NL + yy];
  }
  __syncthreads();

  const int wave = tid >> 5, lane = tid & 31, half = lane >> 4, l16 = lane & 15;
  for (int t = 0; t < 6; ++t) {
    const int tile = wave * 6 + t, tm = tile / 3, tn = tile % 3;
    v8f c; for (int r = 0; r < 8; ++r) c[r] = 0.f;
    for (int kb = 0; kb < NL; kb += 4) {
      const int ka = kb + 2 * half;
      v2f a, bb;
      a.x = Xl[tm * 16 + l16][ka];   a.y = Xl[tm * 16 + l16][ka + 1];
      bb.x = Al[tn * 16 + l16][ka];  bb.y = Al[tn * 16 + l16][ka + 1]; // B = A^T
      c = wmma4(a, bb, c);
    }
    for (int r = 0; r < 8; ++r) {
      const int m = tm * 16 + r + 8 * half;     // input channel
      const int xc = tn * 16 + l16;             // position along line
      const size_t o = ((size_t)b * NC + m) * NSP + n0 + (size_t)xc * stride;
      if (accumulate) s[o] += c[r]; else s[o] = c[r];
    }
  }
}

// ---------------------------------------------------------------------------
// Kernel 5: out = gamma * (Wv * s + bv) + x    (256x256 GEMM, K chunked by 64)
// ---------------------------------------------------------------------------
__global__ __launch_bounds__(256) void kout(
    const float* __restrict__ s, const float* __restrict__ Wv,
    const float* __restrict__ bv, const float* __restrict__ gamma,
    const float* __restrict__ x, float* __restrict__ out)
{
  __shared__ float Wl[256][68];   // pad 68: stride%64==4 banks, 16B aligned
  __shared__ float Sl[64][64];
  __shared__ float bl[256];

  const int b  = blockIdx.y;
  const int n0 = blockIdx.x * 64;
  const int tid = threadIdx.x;
  bl[tid] = bv[tid];
  const float g = gamma[0];

  const int wave = tid >> 5, lane = tid & 31, half = lane >> 4, l16 = lane & 15;
  v8f acc[8];
  for (int t = 0; t < 8; ++t) for (int r = 0; r < 8; ++r) acc[t][r] = 0.f;

  for (int kc = 0; kc < 4; ++kc) {
    __syncthreads();
    for (int i = tid; i < 256 * 16; i += 256) {    // W chunk 256x64
      const int row = i >> 4, col = (i & 15) << 2;
      *(float4*)&Wl[row][col] = *(const float4*)(Wv + row * NC + kc * 64 + col);
    }
    for (int i = tid; i < 64 * 16; i += 256) {     // S chunk 64x64
      const int row = i >> 4, col = (i & 15) << 2;
      *(float4*)&Sl[row][col] =
          *(const float4*)(s + ((size_t)b * NC + kc * 64 + row) * NSP + n0 + col);
    }
    __syncthreads();
    for (int t = 0; t < 8; ++t) {
      const int tile = wave * 8 + t, tm = tile >> 2, tn = tile & 3;
      for (int kb = 0; kb < 64; kb += 4) {
        const int ka = kb + 2 * half;
        v2f a, bb;
        a.x = Wl[tm * 16 + l16][ka];  a.y = Wl[tm * 16 + l16][ka + 1];
        bb.x = Sl[ka][tn * 16 + l16]; bb.y = Sl[ka + 1][tn * 16 + l16];
        acc[t] = wmma4(a, bb, acc[t]);
      }
    }
  }
  for (int t = 0; t < 8; ++t) {
    const int tile = wave * 8 + t, tm = tile >> 2, tn = tile & 3;
    for (int r = 0; r < 8; ++r) {
      const int m = tm * 16 + r + 8 * half;
      const size_t o = ((size_t)b * NC + m) * NSP + n0 + tn * 16 + l16;
      out[o] = g * (acc[t][r] + bl[m]) + x[o];
    }
  }
}

// ---------------------------------------------------------------------------
extern "C" void kernel_launch(void* const* d_in, const int* in_sizes, int n_in,
                              void* d_out, int out_size, void* d_ws, size_t ws_size,
                              hipStream_t stream) {
  const float* x  = (const float*)d_in[0];
  const float* Wq = (const float*)d_in[1];
  const float* bq = (const float*)d_in[2];
  const float* Wk = (const float*)d_in[3];
  const float* bk = (const float*)d_in[4];
  const float* Wv = (const float*)d_in[5];
  const float* bv = (const float*)d_in[6];
  const float* gm = (const float*)d_in[7];
  float* out = (float*)d_out;

  // workspace layout (floats): q | k | att | s
  float* ws  = (float*)d_ws;
  float* q   = ws;                                   // 2*32*110592
  float* k   = ws + (size_t)NB * NE * NSP;           // +7,077,888
  float* att = k  + (size_t)NB * NE * NSP;           // 2*110592*144
  float* s   = att + (size_t)NB * NSP * NATT;        // 2*256*110592

  kproj_qk<<<dim3(NSP / 64, NB), 256, 0, stream>>>(x, Wq, bq, Wk, bk, q, k);
  klogits <<<dim3(NWD, NB, 3),    96, 0, stream>>>(q, k, att);
  ksoftmax<<<dim3(NB * NSP / 8), 256, 0, stream>>>(att, NB * NSP);
  kaggr   <<<dim3(NWD, NB),      256, 0, stream>>>(x, att, s, 0, 0);
  kaggr   <<<dim3(NWD, NB),      256, 0, stream>>>(x, att, s, 1, 1);
  kaggr   <<<dim3(NWD, NB),      256, 0, stream>>>(x, att, s, 2, 1);
  kout    <<<dim3(NSP / 64, NB), 256, 0, stream>>>(s, Wv, bv, gm, x, out);
}